// CTCLoss_50706383896590
// MI455X (gfx1250) — compile-verified
//
#include <hip/hip_runtime.h>
#include <math.h>

// RankNet-style listwise loss:
//   out[b] = logaddexp(0, logsumexp_{i,j}( (s_i - s_j - (1-mask)*1e12) * 20 ))
// factorized to O(N):  lse = M0 + M1 + log(S0) + log(S1)
//   M0 = max_{label=0} 20*s,  S0 = sum_{label=0} exp(20*s - M0)
//   M1 = max_{label=1} -20*s, S1 = sum_{label=1} exp(-20*s - M1)

typedef __attribute__((ext_vector_type(2))) float v2f;
typedef __attribute__((ext_vector_type(8))) float v8f;

#define LSCALE 20.0f

__device__ __forceinline__ float wave_max32(float x) {
#pragma unroll
  for (int off = 16; off >= 1; off >>= 1)
    x = fmaxf(x, __shfl_xor(x, off, 32));
  return x;
}

// Sum x over the 32 lanes of a wave using V_WMMA_F32_16X16X4_F32.
// A (16x4, 2 VGPRs): a[0] = x  -> A[m,0] = x[m] (lanes 0-15), A[m,2] = x[m+16]
//                    a[1] = 0  -> A[m,1] = A[m,3] = 0
// B (4x16, 2 VGPRs): all ones.
// D[m,n] = x[m] + x[m+16] for every n. C/D layout: lane L<16 holds rows 0..7
// in d[0..7]; lane L>=16 holds rows 8..15. Summing the 8 D registers per lane
// then one xor-16 shuffle-add yields the full 32-lane sum in every lane.
__device__ __forceinline__ float wave_sum_wmma(float x) {
  v2f a;    a.x = x;    a.y = 0.0f;
  v2f ones; ones.x = 1.0f; ones.y = 1.0f;
  v8f c = {};
  v8f d = __builtin_amdgcn_wmma_f32_16x16x4_f32(
      /*neg_a=*/false, a, /*neg_b=*/false, ones,
      /*c_mod=*/(short)0, c, /*reuse_a=*/false, /*reuse_b=*/false);
  float s = d[0] + d[1] + d[2] + d[3] + d[4] + d[5] + d[6] + d[7];
  s += __shfl_xor(s, 16, 32);
  return s;
}

__global__ __launch_bounds__(256) void pairwise_lse_kernel(
    const float* __restrict__ scores, const float* __restrict__ labels,
    float* __restrict__ out, int N) {
  const int b    = blockIdx.x;
  const int t    = threadIdx.x;
  const int lane = t & 31;
  const int wid  = t >> 5;
  const int nvec = N >> 2;  // float4 chunks per row

  const float4* s4 = reinterpret_cast<const float4*>(scores) + (size_t)b * nvec;
  const float4* l4 = reinterpret_cast<const float4*>(labels) + (size_t)b * nvec;

  __shared__ float sm0[8], sm1[8], ss0[8], ss1[8];

  // ---- pass 1: per-class maxes (stabilizers) ----
  float m0 = -INFINITY, m1 = -INFINITY;
  for (int i = t; i < nvec; i += 256) {
    float4 s = s4[i];
    float4 l = l4[i];
    float sv[4] = {s.x, s.y, s.z, s.w};
    float lv[4] = {l.x, l.y, l.z, l.w};
#pragma unroll
    for (int k = 0; k < 4; ++k) {
      float z = LSCALE * sv[k];
      if (lv[k] < 0.5f) m0 = fmaxf(m0, z);
      else              m1 = fmaxf(m1, -z);
    }
  }
  m0 = wave_max32(m0);
  m1 = wave_max32(m1);
  if (lane == 0) { sm0[wid] = m0; sm1[wid] = m1; }
  __syncthreads();
  float M0 = -INFINITY, M1 = -INFINITY;
#pragma unroll
  for (int w = 0; w < 8; ++w) {
    M0 = fmaxf(M0, sm0[w]);
    M1 = fmaxf(M1, sm1[w]);
  }

  // ---- pass 2: stabilized sums of exponentials (re-read hits L0/L2) ----
  float p0 = 0.0f, p1 = 0.0f;
  for (int i = t; i < nvec; i += 256) {
    float4 s = s4[i];
    float4 l = l4[i];
    float sv[4] = {s.x, s.y, s.z, s.w};
    float lv[4] = {l.x, l.y, l.z, l.w};
#pragma unroll
    for (int k = 0; k < 4; ++k) {
      float z = LSCALE * sv[k];
      if (lv[k] < 0.5f) p0 += expf(z - M0);
      else              p1 += expf(-z - M1);
    }
  }
  // WMMA-based full-wave reductions (EXEC all ones here: no divergence)
  p0 = wave_sum_wmma(p0);
  p1 = wave_sum_wmma(p1);
  if (lane == 0) { ss0[wid] = p0; ss1[wid] = p1; }
  __syncthreads();

  if (t == 0) {
    float S0 = 0.0f, S1 = 0.0f;
#pragma unroll
    for (int w = 0; w < 8; ++w) { S0 += ss0[w]; S1 += ss1[w]; }
    float lse = (S0 > 0.0f && S1 > 0.0f)
                    ? (M0 + M1 + logf(S0) + logf(S1))
                    : -INFINITY;
    // logaddexp(0, lse), stable; lse = -inf -> 0, matching the fully-masked case
    float r = fmaxf(lse, 0.0f) + log1pf(expf(-fabsf(lse)));
    out[b] = r;
  }
}

extern "C" void kernel_launch(void* const* d_in, const int* in_sizes, int n_in,
                              void* d_out, int out_size, void* d_ws, size_t ws_size,
                              hipStream_t stream) {
  const float* scores = (const float*)d_in[0];
  const float* labels = (const float*)d_in[1];
  float* out = (float*)d_out;

  const int B = out_size;                 // 32
  const int N = in_sizes[0] / (B > 0 ? B : 1);  // 2048

  pairwise_lse_kernel<<<B, 256, 0, stream>>>(scores, labels, out, N);
}